// WhiteAdd_28406913696453
// MI455X (gfx1250) — compile-verified
//
#include <hip/hip_runtime.h>

// WhiteAdd: out[n,c] = codebook[c][ white_table[c][ left[n,c], right[n,c] ] ]
// N = 262144 rows, C = 256 channels, K = 256 codes.
//
// Memory-system-bound kernel for MI455X (23.3 TB/s HBM, 192MB L2):
//  - left/right/out are streamed non-temporally (don't pollute L2)
//  - 64 MiB white_table gathers hit L2 (RT hint)
//  - per-block 64-channel codebook tile staged into LDS via async
//    global->LDS b128 loads (ASYNCcnt), gathered with ds_load_b32.

#define CCH   256   // channels
#define KCODE 256   // code range
#define CT    64    // channel tile per block  -> 64KB LDS (5 blocks/WGP)
#define TPB   256   // threads per block (8 waves, wave32)
#define ROWS_PER_IT   (TPB / CT)                // 4 rows per iteration
#define ROW_ITERS     8
#define ROWS_PER_BLK  (ROWS_PER_IT * ROW_ITERS) // 32 rows per block
#define STAGE_CHUNKS  ((CT * KCODE) / 4 / TPB)  // 16 float4s per thread

// Typed pointers matching the async-to-LDS builtin's signature (AS1/AS3 int4*)
typedef int v4i_cdna5 __attribute__((ext_vector_type(4)));
typedef __attribute__((address_space(1))) v4i_cdna5* g_v4i_p;
typedef __attribute__((address_space(3))) v4i_cdna5* l_v4i_p;

__global__ __launch_bounds__(TPB) void whiteadd_gather_kernel(
    const int* __restrict__ left,
    const int* __restrict__ right,
    const int* __restrict__ wt,        // [C, K, K] int32
    const float* __restrict__ cb,      // [C, K] float32
    float* __restrict__ out,
    int N)
{
    __shared__ __align__(16) float lds_cb[CT * KCODE];   // 64 KB

    const int tid = threadIdx.x;
    const int c0  = blockIdx.y * CT;

    // ---- Stage codebook[c0 : c0+CT, :] into LDS --------------------------
    // 64*256 floats = 4096 float4 chunks; 256 threads x 16 chunks each,
    // fully unrolled straight-line async issues (ASYNCcnt = 16 max).
    {
        const float* gsrc = cb + (size_t)c0 * KCODE + (size_t)tid * 4;
        float*       ldst = &lds_cb[tid * 4];
#if __has_builtin(__builtin_amdgcn_global_load_async_to_lds_b128)
#pragma unroll
        for (int k = 0; k < STAGE_CHUNKS; ++k) {
            __builtin_amdgcn_global_load_async_to_lds_b128(
                (g_v4i_p)(gsrc + (size_t)k * TPB * 4),
                (l_v4i_p)(ldst + k * TPB * 4),
                /*offset=*/0, /*cpol=*/0);
        }
  #if __has_builtin(__builtin_amdgcn_s_wait_asynccnt)
        __builtin_amdgcn_s_wait_asynccnt(0);
  #else
        asm volatile("s_wait_asynccnt 0" ::: "memory");
  #endif
#else
        // Fallback: coalesced b128 load + ds_store_b128
#pragma unroll
        for (int k = 0; k < STAGE_CHUNKS; ++k) {
            *(float4*)(ldst + k * TPB * 4) = *(const float4*)(gsrc + (size_t)k * TPB * 4);
        }
#endif
    }
    __syncthreads();

    // ---- Main double-gather loop ----------------------------------------
    const int c_local = tid & (CT - 1);      // lanes -> contiguous channels
    const int r_off   = tid >> 6;            // row within iteration group
    const int c       = c0 + c_local;
    const int n_base  = blockIdx.x * ROWS_PER_BLK + r_off;

    const int   wt_c_base = c << 16;         // c * K * K
    const float* cb_loc   = &lds_cb[c_local * KCODE];

    if ((unsigned)(blockIdx.x + 1) * ROWS_PER_BLK <= (unsigned)N) {
        // Fast path: fully in range, deep unroll for memory-level parallelism.
#pragma unroll
        for (int it = 0; it < ROW_ITERS; ++it) {
            const int n = n_base + it * ROWS_PER_IT;
            const size_t off = (size_t)n * CCH + c;
            // Streaming: NT so these 768 MiB don't evict the L2-resident table
            const int l = __builtin_nontemporal_load(left + off);
            const int r = __builtin_nontemporal_load(right + off);
            // Table gather: RT (default) -> stays in 192MB L2 (table = 64MB)
            const int code = wt[wt_c_base + (l << 8) + r];
            // Codebook gather from LDS
            const float v = cb_loc[code];
            __builtin_nontemporal_store(v, out + off);
        }
    } else {
        for (int it = 0; it < ROW_ITERS; ++it) {
            const int n = n_base + it * ROWS_PER_IT;
            if (n >= N) break;
            const size_t off = (size_t)n * CCH + c;
            const int l = __builtin_nontemporal_load(left + off);
            const int r = __builtin_nontemporal_load(right + off);
            const int code = wt[wt_c_base + (l << 8) + r];
            const float v = cb_loc[code];
            __builtin_nontemporal_store(v, out + off);
        }
    }
}

extern "C" void kernel_launch(void* const* d_in, const int* in_sizes, int n_in,
                              void* d_out, int out_size, void* d_ws, size_t ws_size,
                              hipStream_t stream) {
    const int*   left  = (const int*)d_in[0];
    const int*   right = (const int*)d_in[1];
    const int*   wt    = (const int*)d_in[2];
    const float* cb    = (const float*)d_in[3];
    float*       out   = (float*)d_out;

    const int N = in_sizes[0] / CCH;   // rows

    dim3 grid((N + ROWS_PER_BLK - 1) / ROWS_PER_BLK, CCH / CT);
    whiteadd_gather_kernel<<<grid, TPB, 0, stream>>>(left, right, wt, cb, out, N);
}